// KVCache_51161650430103
// MI455X (gfx1250) — compile-verified
//
#include <hip/hip_runtime.h>

typedef __attribute__((ext_vector_type(2))) float v2f;
typedef __attribute__((ext_vector_type(8))) float v8f;

#define B_   8
#define T_   4096
#define S_   4096
#define HL_  8
#define D_   128
#define BS_  64
#define TB_  64

// Flat output offsets (concat order of the reference tuple, all as f32)
static constexpr size_t OFF_KC   = 0;                               // k_cache   [8,4096,8,128]
static constexpr size_t OFF_VC   = 33554432;                        // v_cache   [8,4096,8,128]
static constexpr size_t OFF_VNT  = 67108864;                        // v_norm_tok[8,4096,8]
static constexpr size_t OFF_KSUM = 67371008;                        // k_sum_blk [8,64,8,128]
static constexpr size_t OFF_KCNT = 67895296;                        // k_cnt_blk [8,64]
static constexpr size_t OFF_VNB  = 67895808;                        // v_norm_blk[8,64,8]
static constexpr size_t OFF_PRE  = 67899904;                        // prefill_len scalar

// ---------------------------------------------------------------------------
// Kernel 1: token scatter. One 256-thread block per (b, s) row; a row is
// Hl*D = 1024 floats = 256 float4, so each thread moves one float4 for k and
// one for v (global_load_b128 / global_store_b128 — pure streaming).
// ---------------------------------------------------------------------------
__global__ __launch_bounds__(256) void scatter_copy_kernel(
    const float* __restrict__ k_bshd, const float* __restrict__ v_bshd,
    const float* __restrict__ v_norm_bsh, const int* __restrict__ pos,
    float* __restrict__ out)
{
  const int bs_idx = blockIdx.x;          // b*S + s
  const int b = bs_idx >> 12;             // / 4096
  const int s = bs_idx & (S_ - 1);
  const int p = pos[s];                   // scalar-cached broadcast

  const size_t src = (size_t)bs_idx * (HL_ * D_);
  const size_t dst = ((size_t)b * T_ + (size_t)p) * (HL_ * D_);

  const float4* __restrict__ ks = (const float4*)(k_bshd + src);
  const float4* __restrict__ vs = (const float4*)(v_bshd + src);
  float4* kd = (float4*)(out + OFF_KC + dst);
  float4* vd = (float4*)(out + OFF_VC + dst);

  const int t = threadIdx.x;
  kd[t] = ks[t];
  vd[t] = vs[t];

  if (t < HL_) {
    out[OFF_VNT + ((size_t)b * T_ + (size_t)p) * HL_ + t] =
        v_norm_bsh[(size_t)bs_idx * HL_ + t];
  }
}

// ---------------------------------------------------------------------------
// Kernel 2: per-block reductions. One wave32 per (b, blk, h) combo; 8 waves
// per 256-thread block.  Block key-sum via V_WMMA_F32_16X16X4_F32:
//   A (16x4 f32)  = K[dims dbase..dbase+15 ; 4 tokens]   (ISA A layout)
//   B (4x16 f32)  = all ones
//   C += A*B  -> D[m, n] = sum over tokens of K[t, dbase+m]  (all n equal)
// 16 accumulation steps cover the 64 tokens of the block.
// ---------------------------------------------------------------------------
__global__ __launch_bounds__(256) void block_reduce_wmma_kernel(
    const float* __restrict__ k_bshd, const float* __restrict__ v_norm_bsh,
    const float* __restrict__ k_sum_in, const int* __restrict__ k_cnt_in,
    const float* __restrict__ v_norm_blk_in, float* __restrict__ out)
{
  const int lane  = threadIdx.x & 31;
  const int wave  = threadIdx.x >> 5;
  const int combo = blockIdx.x * 8 + wave;     // [0, B*Tb*Hl) = [0, 4096)
  const int h   = combo & (HL_ - 1);
  const int blk = (combo >> 3) & (TB_ - 1);
  const int b   = combo >> 9;

  // WMMA 32-bit A-matrix 16x4 layout: lanes 0-15 = M rows with K=0 (VGPR0),
  // K=1 (VGPR1); lanes 16-31 = same M rows with K=2, K=3.
  const int half = lane >> 4;                  // 0: tokens +0/+1, 1: +2/+3
  const int m    = lane & 15;                  // dim row within 16-wide group

  const size_t tokstride = (size_t)HL_ * D_;   // 1024 floats between tokens
  const float* kbase =
      k_bshd + (((size_t)b * S_ + (size_t)blk * BS_) * HL_ + h) * D_;

  v2f ones; ones.x = 1.0f; ones.y = 1.0f;

  for (int dbase = 0; dbase < D_; dbase += 16) {
    v8f c = {0.f, 0.f, 0.f, 0.f, 0.f, 0.f, 0.f, 0.f};
    const float* pa = kbase + dbase + m + (size_t)(2 * half) * tokstride;
    for (int it = 0; it < 16; ++it) {
      v2f a;
      a.x = pa[0];                 // token 4*it + 2*half
      a.y = pa[tokstride];         // token 4*it + 2*half + 1
      c = __builtin_amdgcn_wmma_f32_16x16x4_f32(
              false, a, false, ones, (short)0, c, false, false);
      pa += 4 * tokstride;
    }
    // C/D layout: VGPR r, lanes 0-15 -> M=r; lanes 16-31 -> M=8+r.
    if (m == 0) {
      const size_t ksbase =
          (((size_t)b * TB_ + blk) * HL_ + h) * D_ + dbase + half * 8;
      #pragma unroll
      for (int r = 0; r < 8; ++r)
        out[OFF_KSUM + ksbase + r] = k_sum_in[ksbase + r] + c[r];
    }
  }

  // Per-block running max of v-norms for this head (64 tokens / 32 lanes).
  const float* vn =
      v_norm_bsh + ((size_t)b * S_ + (size_t)blk * BS_) * HL_ + h;
  float mv = fmaxf(vn[(size_t)lane * HL_], vn[(size_t)(lane + 32) * HL_]);
  #pragma unroll
  for (int mask = 16; mask >= 1; mask >>= 1)
    mv = fmaxf(mv, __shfl_xor(mv, mask, 32));

  if (lane == 0) {
    const size_t i = ((size_t)b * TB_ + blk) * HL_ + h;
    out[OFF_VNB + i] = fmaxf(v_norm_blk_in[i], mv);
    if (h == 0) {
      const size_t j = (size_t)b * TB_ + blk;
      out[OFF_KCNT + j] = (float)(k_cnt_in[j] + BS_);
    }
  }
}

// ---------------------------------------------------------------------------
// Kernel 3: prefill_len = max(input_pos_s) + 1 (tiny reduction over 4096 ints)
// ---------------------------------------------------------------------------
__global__ __launch_bounds__(256) void prefill_kernel(
    const int* __restrict__ pos, float* __restrict__ out)
{
  __shared__ int sm[256];
  const int t = threadIdx.x;
  int mx = -2147483647 - 1;
  for (int i = t; i < S_; i += 256) mx = max(mx, pos[i]);
  sm[t] = mx;
  __syncthreads();
  for (int off = 128; off >= 1; off >>= 1) {
    if (t < off) sm[t] = max(sm[t], sm[t + off]);
    __syncthreads();
  }
  if (t == 0) out[OFF_PRE] = (float)(sm[0] + 1);
}

// ---------------------------------------------------------------------------
extern "C" void kernel_launch(void* const* d_in, const int* in_sizes, int n_in,
                              void* d_out, int out_size, void* d_ws, size_t ws_size,
                              hipStream_t stream) {
  (void)in_sizes; (void)n_in; (void)out_size; (void)d_ws; (void)ws_size;

  const float* k_sum_in      = (const float*)d_in[3];
  const int*   k_cnt_in      = (const int*)  d_in[4];
  const float* v_norm_blk_in = (const float*)d_in[5];
  const int*   pos           = (const int*)  d_in[6];
  const float* k_bshd        = (const float*)d_in[7];
  const float* v_bshd        = (const float*)d_in[8];
  const float* v_norm_bsh    = (const float*)d_in[9];
  float* out = (float*)d_out;

  // Scatter / cache overwrite: B*S workgroups of 256 threads.
  scatter_copy_kernel<<<B_ * S_, 256, 0, stream>>>(
      k_bshd, v_bshd, v_norm_bsh, pos, out);

  // Block reductions: 4096 wave-combos / 8 waves per block = 512 blocks.
  block_reduce_wmma_kernel<<<(B_ * TB_ * HL_) / 8, 256, 0, stream>>>(
      k_bshd, v_norm_bsh, k_sum_in, k_cnt_in, v_norm_blk_in, out);

  prefill_kernel<<<1, 256, 0, stream>>>(pos, out);
}